// GATClassifier_55259049230616
// MI455X (gfx1250) — compile-verified
//
#include <hip/hip_runtime.h>
#include <math.h>

#define NN    50000
#define EE    800000
#define FDIM  128      // HEADS*HID
#define NH    4        // heads
#define HD    32       // hidden per head
#define NCLS  10

typedef float v2f __attribute__((ext_vector_type(2)));
typedef float v8f __attribute__((ext_vector_type(8)));

// ---------------- float atomic max via sign-split integer atomics ----------
__device__ __forceinline__ void atomic_max_f32(float* addr, float val) {
  if (val >= 0.0f) atomicMax((int*)addr, __float_as_int(val));
  else             atomicMin((unsigned int*)addr, __float_as_uint(val));
}

// ---------------- C[M,128] = A[M,128] x B[128,128] via V_WMMA_F32_16X16X4_F32
// One wave per 16x16 output tile; 8 waves/block cover the 8 column tiles of a
// 16-row stripe (A rows reused out of L0 across the block's waves).
// ISA layouts (CDNA5 ISA):
//   A 16x4 f32 : lane&15 = M ; lanes>=16 hold K+2 ; vgpr0=K, vgpr1=K+1
//   B 4x16 f32 : lane&15 = N ; lanes>=16 hold K+2 ; vgpr0=K, vgpr1=K+1
//   C/D 16x16  : lane&15 = N ; vgpr v = row v (+8 for lanes>=16)
__global__ __launch_bounds__(256)
void gemm128_wmma(const float* __restrict__ A, const float* __restrict__ B,
                  float* __restrict__ C) {
  const int lane    = threadIdx.x & 31;
  const int wave    = threadIdx.x >> 5;
  const int rowBase = blockIdx.x * 16;
  const int colBase = wave * 16;
  const int l15     = lane & 15;
  const int kh      = (lane >> 4) * 2;       // 0 or 2
  const float* Arow = A + (size_t)(rowBase + l15) * FDIM;

  v8f acc = {0.f, 0.f, 0.f, 0.f, 0.f, 0.f, 0.f, 0.f};
#pragma unroll
  for (int k = 0; k < FDIM; k += 4) {
    v2f a = *(const v2f*)(Arow + k + kh);            // A[m][k+kh], A[m][k+kh+1]
    v2f b;
    b.x = B[(size_t)(k + kh    ) * FDIM + colBase + l15];
    b.y = B[(size_t)(k + kh + 1) * FDIM + colBase + l15];
    acc = __builtin_amdgcn_wmma_f32_16x16x4_f32(false, a, false, b,
                                                (short)0, acc, false, false);
  }
  const int rOff = (lane >> 4) * 8;
#pragma unroll
  for (int v = 0; v < 8; ++v)
    C[(size_t)(rowBase + rOff + v) * FDIM + colBase + l15] = acc[v];
}

// ---------------- classifier: Out[M,10] = A[M,128] x Wc[128,10] + bc --------
__global__ __launch_bounds__(256)
void gemm_cls_wmma(const float* __restrict__ A, const float* __restrict__ Wc,
                   const float* __restrict__ bc, float* __restrict__ Out) {
  const int lane    = threadIdx.x & 31;
  const int wave    = threadIdx.x >> 5;
  const int rowTile = blockIdx.x * 8 + wave;
  if (rowTile >= NN / 16) return;                    // whole-wave exit
  const int rowBase = rowTile * 16;
  const int l15     = lane & 15;
  const int kh      = (lane >> 4) * 2;
  const float* Arow = A + (size_t)(rowBase + l15) * FDIM;

  v8f acc = {0.f, 0.f, 0.f, 0.f, 0.f, 0.f, 0.f, 0.f};
#pragma unroll
  for (int k = 0; k < FDIM; k += 4) {
    v2f a = *(const v2f*)(Arow + k + kh);
    v2f b;
    b.x = (l15 < NCLS) ? Wc[(size_t)(k + kh    ) * NCLS + l15] : 0.f;
    b.y = (l15 < NCLS) ? Wc[(size_t)(k + kh + 1) * NCLS + l15] : 0.f;
    acc = __builtin_amdgcn_wmma_f32_16x16x4_f32(false, a, false, b,
                                                (short)0, acc, false, false);
  }
  if (l15 < NCLS) {
    const int   rOff = (lane >> 4) * 8;
    const float bias = bc[l15];
#pragma unroll
    for (int v = 0; v < 8; ++v)
      Out[(size_t)(rowBase + rOff + v) * NCLS + l15] = acc[v] + bias;
  }
}

// ---------------- attention logits: el/er[n,h] = <h[n,h,:], a_l/r[h,:]> -----
__global__ __launch_bounds__(256)
void elr_kernel(const float* __restrict__ h, const float* __restrict__ al,
                const float* __restrict__ ar, float* __restrict__ el,
                float* __restrict__ er) {
  const int idx = blockIdx.x * blockDim.x + threadIdx.x;
  if (idx >= NN * NH) return;
  const int n = idx >> 2, hd = idx & 3;
  const float* hp  = h  + (size_t)n * FDIM + hd * HD;
  const float* alp = al + hd * HD;
  const float* arp = ar + hd * HD;
  float sl = 0.f, sr = 0.f;
#pragma unroll
  for (int d = 0; d < HD; ++d) { float v = hp[d]; sl += v * alp[d]; sr += v * arp[d]; }
  el[idx] = sl;
  er[idx] = sr;
}

// ---------------- per-layer state init ------------------------------------
__global__ __launch_bounds__(256)
void init_kernel(float* __restrict__ agg, float* __restrict__ m,
                 float* __restrict__ denom) {
  const int idx = blockIdx.x * blockDim.x + threadIdx.x;
  if (idx < NN * FDIM) agg[idx] = 0.f;
  if (idx < NN * NH) { m[idx] = -__builtin_inff(); denom[idx] = 0.f; }
}

// ---------------- edge passes ---------------------------------------------
__global__ __launch_bounds__(256)
void edge_score_kernel(const int* __restrict__ src, const int* __restrict__ dst,
                       const float* __restrict__ el, const float* __restrict__ er,
                       float* __restrict__ eb, float* __restrict__ m) {
  const int idx = blockIdx.x * blockDim.x + threadIdx.x;
  if (idx >= EE * NH) return;
  const int e = idx >> 2, hd = idx & 3;
  const int s = src[e], d = dst[e];
  float v = el[s * NH + hd] + er[d * NH + hd];
  v = (v > 0.f) ? v : 0.2f * v;                      // leaky_relu, slope 0.2
  eb[idx] = v;
  atomic_max_f32(&m[d * NH + hd], v);
}

__global__ __launch_bounds__(256)
void edge_exp_kernel(const int* __restrict__ dst, float* __restrict__ eb,
                     const float* __restrict__ m, float* __restrict__ denom) {
  const int idx = blockIdx.x * blockDim.x + threadIdx.x;
  if (idx >= EE * NH) return;
  const int e = idx >> 2, hd = idx & 3;
  const int d = dst[e];
  const float ex = __expf(eb[idx] - m[d * NH + hd]);
  eb[idx] = ex;
  atomicAdd(&denom[d * NH + hd], ex);
}

__global__ __launch_bounds__(256)
void edge_alpha_kernel(const int* __restrict__ dst, float* __restrict__ eb,
                       const float* __restrict__ denom) {
  const int idx = blockIdx.x * blockDim.x + threadIdx.x;
  if (idx >= EE * NH) return;
  const int e = idx >> 2, hd = idx & 3;
  eb[idx] = eb[idx] / fmaxf(denom[dst[e] * NH + hd], 1e-20f);
}

// Heavy kernel: each wave32 covers 32 consecutive channels of one (edge,head)
// -> wave-uniform alpha/src/dst, 128B-coalesced gather + L2 atomic scatter.
__global__ __launch_bounds__(256)
void edge_agg_kernel(const int* __restrict__ src, const int* __restrict__ dst,
                     const float* __restrict__ h, const float* __restrict__ alpha,
                     float* __restrict__ agg) {
  const int t = blockIdx.x * blockDim.x + threadIdx.x;   // < EE*128 (~102.4M)
  if (t >= EE * FDIM) return;
  const int e  = t >> 7;
  const int c  = t & 127;
  const int hd = c >> 5;
  const float a = alpha[e * NH + hd];
  const int s = src[e], d = dst[e];
  atomicAdd(&agg[(size_t)d * FDIM + c], a * h[(size_t)s * FDIM + c]);
}

__global__ __launch_bounds__(256)
void bias_relu_kernel(float* __restrict__ x, const float* __restrict__ b) {
  const int idx = blockIdx.x * blockDim.x + threadIdx.x;
  if (idx >= NN * FDIM) return;
  x[idx] = fmaxf(x[idx] + b[idx & 127], 0.f);
}

// ---------------------------------------------------------------------------
extern "C" void kernel_launch(void* const* d_in, const int* in_sizes, int n_in,
                              void* d_out, int out_size, void* d_ws, size_t ws_size,
                              hipStream_t stream) {
  const float* features = (const float*)d_in[0];
  const int*   src      = (const int*)d_in[1];
  const int*   dst      = (const int*)d_in[2];
  const float* W1  = (const float*)d_in[3];
  const float* al1 = (const float*)d_in[4];
  const float* ar1 = (const float*)d_in[5];
  const float* b1  = (const float*)d_in[6];
  const float* W2  = (const float*)d_in[7];
  const float* al2 = (const float*)d_in[8];
  const float* ar2 = (const float*)d_in[9];
  const float* b2  = (const float*)d_in[10];
  const float* Wc  = (const float*)d_in[11];
  const float* bc  = (const float*)d_in[12];
  float* out = (float*)d_out;

  // workspace layout (floats)
  const size_t HZ   = (size_t)NN * FDIM;   // 6.4M
  const size_t EB   = (size_t)EE * NH;     // 3.2M
  const size_t NHsz = (size_t)NN * NH;     // 0.2M
  const size_t need = (2 * HZ + EB + 4 * NHsz) * sizeof(float);  // ~67.2 MB
  if (ws_size < need) return;

  float* ws    = (float*)d_ws;
  float* hbuf  = ws;             // [N,128] post-GEMM features
  float* agg   = ws + HZ;        // [N,128] aggregation accumulator / layer input
  float* eb    = ws + 2 * HZ;    // [E,4]   e -> ex -> alpha
  float* el    = eb + EB;
  float* er    = el + NHsz;
  float* m     = er + NHsz;
  float* denom = m  + NHsz;

  const int T = 256;
  const int gNH  = (NN * NH + T - 1) / T;
  const int gHZ  = (NN * FDIM + T - 1) / T;
  const int gEB  = (EE * NH + T - 1) / T;
  const int gAGG = (EE * FDIM + T - 1) / T;
  const int gGEMM = NN / 16;                 // 3125 (exact)
  const int gCLS  = (NN / 16 + 7) / 8;

  // ---------------- layer 1 ----------------
  gemm128_wmma<<<gGEMM, T, 0, stream>>>(features, W1, hbuf);
  elr_kernel<<<gNH, T, 0, stream>>>(hbuf, al1, ar1, el, er);
  init_kernel<<<gHZ, T, 0, stream>>>(agg, m, denom);
  edge_score_kernel<<<gEB, T, 0, stream>>>(src, dst, el, er, eb, m);
  edge_exp_kernel<<<gEB, T, 0, stream>>>(dst, eb, m, denom);
  edge_alpha_kernel<<<gEB, T, 0, stream>>>(dst, eb, denom);
  edge_agg_kernel<<<gAGG, T, 0, stream>>>(src, dst, hbuf, eb, agg);
  bias_relu_kernel<<<gHZ, T, 0, stream>>>(agg, b1);

  // ---------------- layer 2 (agg holds x2; GEMM reads it before re-init) ---
  gemm128_wmma<<<gGEMM, T, 0, stream>>>(agg, W2, hbuf);
  elr_kernel<<<gNH, T, 0, stream>>>(hbuf, al2, ar2, el, er);
  init_kernel<<<gHZ, T, 0, stream>>>(agg, m, denom);
  edge_score_kernel<<<gEB, T, 0, stream>>>(src, dst, el, er, eb, m);
  edge_exp_kernel<<<gEB, T, 0, stream>>>(dst, eb, m, denom);
  edge_alpha_kernel<<<gEB, T, 0, stream>>>(dst, eb, denom);
  edge_agg_kernel<<<gAGG, T, 0, stream>>>(src, dst, hbuf, eb, agg);
  bias_relu_kernel<<<gHZ, T, 0, stream>>>(agg, b2);

  // ---------------- classifier ----------------
  gemm_cls_wmma<<<gCLS, T, 0, stream>>>(agg, Wc, bc, out);
}